// PushingEnv_65635690218005
// MI455X (gfx1250) — compile-verified
//
#include <hip/hip_runtime.h>

// Forward simulation of B independent 2-body contact environments, H steps:
//   f   = -K * relu(x1 + 1 - x2)          (K = 100)
//   v1 += (f  - C*v1) * DT                (C = 1, DT = 0.01)
//   v2 += (-f - C*v2) * DT
//   x1 += v1 * DT ; x2 += v2 * DT         (semi-implicit Euler)
// then loss = (x2 - goal)^2. Output [3, B]: row0 = loss, row1 = x1, row2 = x2.
// u_seq / gtype do not affect the forward math.
//
// Since K*DT == 1 exactly, the step reduces to (r = relu(x1 - x2 + 1)):
//   v1 = fma(v1, 0.99, -r); v2 = fma(v2, 0.99, +r)
//   x1 = fma(v1, DT, x1);   x2 = fma(v2, DT, x2)
// = 7 packed ops per step per float2 chain (sub, add, max, 4x fma), with the
// negations folding into VOP3P NEG modifiers. VALU-bound (~63 FLOP/byte), so
// we target packed-fp32 (v_pk_fma_f32 etc.) via float2 ext-vectors, two
// independent chains per thread for ILP, and b128 coalesced VMEM.

typedef float v2f __attribute__((ext_vector_type(2)));

__device__ __forceinline__ v2f relu2(v2f a) {
    const v2f z = {0.0f, 0.0f};
    return __builtin_elementwise_max(a, z);   // maxnum semantics == jnp.maximum
}

__global__ __launch_bounds__(256) void pushing_env_fwd(
    const float* __restrict__ x1_0,
    const float* __restrict__ v1_0,
    const float* __restrict__ x2_0,
    const float* __restrict__ v2_0,
    const float* __restrict__ goal,
    float* __restrict__ out,
    int B, int H)
{
    const float DT = 0.01f;
    const v2f  dtv = {DT, DT};
    const v2f  av  = {1.0f - DT, 1.0f - DT};   // 0.99 damping factor

    // Clamp the last workgroup's threads onto the final full float4 group:
    // duplicated threads recompute identical values and store identical bytes,
    // so this is deterministic and removes the scalar tail entirely.
    // (Requires B >= 4; harness B = 262144, a multiple of 1024.)
    int i = (blockIdx.x * blockDim.x + threadIdx.x) * 4;
    i = min(i, B - 4);

    // 128-bit coalesced loads: 4 environments per thread.
    const float4 X1 = *(const float4*)(x1_0 + i);
    const float4 V1 = *(const float4*)(v1_0 + i);
    const float4 X2 = *(const float4*)(x2_0 + i);
    const float4 V2 = *(const float4*)(v2_0 + i);
    const float4 G  = *(const float4*)(goal + i);

    // Two independent packed-f32 chains (a: envs 0-1, b: envs 2-3).
    v2f x1a = {X1.x, X1.y}, x1b = {X1.z, X1.w};
    v2f v1a = {V1.x, V1.y}, v1b = {V1.z, V1.w};
    v2f x2a = {X2.x, X2.y}, x2b = {X2.z, X2.w};
    v2f v2a = {V2.x, V2.y}, v2b = {V2.z, V2.w};

    #pragma unroll 4
    for (int t = 0; t < H; ++t) {
        // r = relu(x1 - x2 + 1) ; note f*DT = -K*DT*r = -r since K*DT == 1
        v2f ra = relu2((x1a - x2a) + 1.0f);
        v2f rb = relu2((x1b - x2b) + 1.0f);
        // v1' = 0.99*v1 - r ; v2' = 0.99*v2 + r
        v1a = __builtin_elementwise_fma(v1a, av, -ra);
        v1b = __builtin_elementwise_fma(v1b, av, -rb);
        v2a = __builtin_elementwise_fma(v2a, av,  ra);
        v2b = __builtin_elementwise_fma(v2b, av,  rb);
        // x += v' * DT
        x1a = __builtin_elementwise_fma(v1a, dtv, x1a);
        x1b = __builtin_elementwise_fma(v1b, dtv, x1b);
        x2a = __builtin_elementwise_fma(v2a, dtv, x2a);
        x2b = __builtin_elementwise_fma(v2b, dtv, x2b);
    }

    v2f ga = {G.x, G.y}, gb = {G.z, G.w};
    v2f la = x2a - ga;  la = la * la;
    v2f lb = x2b - gb;  lb = lb * lb;

    // 128-bit coalesced stores into the [3, B] output.
    *(float4*)(out + i)         = make_float4(la.x,  la.y,  lb.x,  lb.y);
    *(float4*)(out + B + i)     = make_float4(x1a.x, x1a.y, x1b.x, x1b.y);
    *(float4*)(out + 2 * B + i) = make_float4(x2a.x, x2a.y, x2b.x, x2b.y);
}

extern "C" void kernel_launch(void* const* d_in, const int* in_sizes, int n_in,
                              void* d_out, int out_size, void* d_ws, size_t ws_size,
                              hipStream_t stream) {
    // setup_inputs() order: u_seq[H], x1_0[B], v1_0[B], x2_0[B], v2_0[B],
    // goal[B], gtype[1]. u_seq and gtype are unused by the forward math.
    const int   H    = in_sizes[0];
    const int   B    = in_sizes[1];
    const float* x1_0 = (const float*)d_in[1];
    const float* v1_0 = (const float*)d_in[2];
    const float* x2_0 = (const float*)d_in[3];
    const float* v2_0 = (const float*)d_in[4];
    const float* goal = (const float*)d_in[5];
    float* out = (float*)d_out;

    const int threads = 256;             // 8 wave32 per block
    const int envs_per_thread = 4;
    const int blocks = (B + threads * envs_per_thread - 1) / (threads * envs_per_thread);

    pushing_env_fwd<<<blocks, threads, 0, stream>>>(
        x1_0, v1_0, x2_0, v2_0, goal, out, B, H);
}